// DiscriminativeLoss_81930796138948
// MI455X (gfx1250) — compile-verified
//
#include <hip/hip_runtime.h>
#include <math.h>

#define K_SEG 33
#define NIMG  8
#define EDIM  16
#define EPAD  17            // padded segment-row stride (floats): 17 coprime with 64 banks
#define NPIX  (512*512)

#define DELTA_V 0.5f
#define DELTA_D 1.5f
#define ALPHA   1.0f
#define BETA    1.0f
#define GAMMA   0.001f

typedef __attribute__((ext_vector_type(2))) float v2f;
typedef __attribute__((ext_vector_type(8))) float v8f;

// workspace layout (floats)
#define OFF_COUNTS 0
#define OFF_SUMS   (NIMG*K_SEG)
#define OFF_MEANS  (OFF_SUMS + NIMG*K_SEG*EDIM)
#define OFF_VARPER (OFF_MEANS + NIMG*K_SEG*EDIM)
#define WS_FLOATS  (OFF_VARPER + NIMG*K_SEG)

#define PIX_BLOCKS 128   // blocks per image for streaming passes
#define TPB 256

__global__ void k_zero(float* __restrict__ ws) {
    int i = blockIdx.x * blockDim.x + threadIdx.x;
    if (i < WS_FLOATS) ws[i] = 0.f;
}

// Pass 1: per-segment counts and embedding sums (float4-vectorized streaming).
// LDS segment rows padded to stride 17 so distinct segment ids hit distinct banks.
__global__ void k_segsum(const float* __restrict__ emb, const int* __restrict__ mask,
                         float* __restrict__ ws) {
    __shared__ float s_sum[K_SEG * EPAD];
    __shared__ float s_cnt[K_SEG];
    const int tid = threadIdx.x;
    for (int i = tid; i < K_SEG * EPAD; i += blockDim.x) s_sum[i] = 0.f;
    for (int i = tid; i < K_SEG; i += blockDim.x) s_cnt[i] = 0.f;
    __syncthreads();

    const int img = blockIdx.y;
    const float4* e4 = reinterpret_cast<const float4*>(emb + (size_t)img * EDIM * NPIX);
    const int4*   m4 = reinterpret_cast<const int4*>(mask + (size_t)img * NPIX);

    const int quadsPerBlock = (NPIX / 4) / gridDim.x;
    const int q0 = blockIdx.x * quadsPerBlock;
    for (int q = q0 + tid; q < q0 + quadsPerBlock; q += blockDim.x) {
        int4 id = m4[q];
        const int bx = id.x * EPAD, by = id.y * EPAD, bz = id.z * EPAD, bw = id.w * EPAD;
        atomicAdd(&s_cnt[id.x], 1.f);
        atomicAdd(&s_cnt[id.y], 1.f);
        atomicAdd(&s_cnt[id.z], 1.f);
        atomicAdd(&s_cnt[id.w], 1.f);
        #pragma unroll
        for (int e = 0; e < EDIM; ++e) {
            float4 v = e4[(size_t)e * (NPIX / 4) + q];
            atomicAdd(&s_sum[bx + e], v.x);
            atomicAdd(&s_sum[by + e], v.y);
            atomicAdd(&s_sum[bz + e], v.z);
            atomicAdd(&s_sum[bw + e], v.w);
        }
    }
    __syncthreads();

    float* gcnt = ws + OFF_COUNTS + img * K_SEG;
    float* gsum = ws + OFF_SUMS   + img * K_SEG * EDIM;
    for (int i = tid; i < K_SEG; i += blockDim.x)
        if (s_cnt[i] != 0.f) atomicAdd(&gcnt[i], s_cnt[i]);
    for (int i = tid; i < K_SEG * EDIM; i += blockDim.x) {
        float v = s_sum[(i / EDIM) * EPAD + (i % EDIM)];
        if (v != 0.f) atomicAdd(&gsum[i], v);
    }
}

__global__ void k_means(float* __restrict__ ws) {
    int i = blockIdx.x * blockDim.x + threadIdx.x;
    if (i >= NIMG * K_SEG * EDIM) return;
    int img_k = i / EDIM;
    float c = fmaxf(ws[OFF_COUNTS + img_k], 1.f);
    ws[OFF_MEANS + i] = ws[OFF_SUMS + i] / c;
}

// Pass 2: per-pixel variance hinge term, segment-accumulated.
__global__ void k_var(const float* __restrict__ emb, const int* __restrict__ mask,
                      float* __restrict__ ws) {
    __shared__ float s_mean[K_SEG * EPAD];
    __shared__ float s_var[K_SEG];
    const int tid = threadIdx.x;
    const int img = blockIdx.y;
    for (int i = tid; i < K_SEG * EDIM; i += blockDim.x)
        s_mean[(i / EDIM) * EPAD + (i % EDIM)] =
            ws[OFF_MEANS + img * K_SEG * EDIM + i];
    for (int i = tid; i < K_SEG; i += blockDim.x) s_var[i] = 0.f;
    __syncthreads();

    const float4* e4 = reinterpret_cast<const float4*>(emb + (size_t)img * EDIM * NPIX);
    const int4*   m4 = reinterpret_cast<const int4*>(mask + (size_t)img * NPIX);

    const int quadsPerBlock = (NPIX / 4) / gridDim.x;
    const int q0 = blockIdx.x * quadsPerBlock;
    for (int q = q0 + tid; q < q0 + quadsPerBlock; q += blockDim.x) {
        int4 id = m4[q];
        const int bx = id.x * EPAD, by = id.y * EPAD, bz = id.z * EPAD, bw = id.w * EPAD;
        float d0 = 0.f, d1 = 0.f, d2 = 0.f, d3 = 0.f;
        #pragma unroll
        for (int e = 0; e < EDIM; ++e) {
            float4 v = e4[(size_t)e * (NPIX / 4) + q];
            float t;
            t = v.x - s_mean[bx + e]; d0 = fmaf(t, t, d0);
            t = v.y - s_mean[by + e]; d1 = fmaf(t, t, d1);
            t = v.z - s_mean[bz + e]; d2 = fmaf(t, t, d2);
            t = v.w - s_mean[bw + e]; d3 = fmaf(t, t, d3);
        }
        float h;
        if (id.x) { h = fmaxf((d0 > 0.f ? sqrtf(d0) : 0.f) - DELTA_V, 0.f); if (h != 0.f) atomicAdd(&s_var[id.x], h * h); }
        if (id.y) { h = fmaxf((d1 > 0.f ? sqrtf(d1) : 0.f) - DELTA_V, 0.f); if (h != 0.f) atomicAdd(&s_var[id.y], h * h); }
        if (id.z) { h = fmaxf((d2 > 0.f ? sqrtf(d2) : 0.f) - DELTA_V, 0.f); if (h != 0.f) atomicAdd(&s_var[id.z], h * h); }
        if (id.w) { h = fmaxf((d3 > 0.f ? sqrtf(d3) : 0.f) - DELTA_V, 0.f); if (h != 0.f) atomicAdd(&s_var[id.w], h * h); }
    }
    __syncthreads();

    float* gvar = ws + OFF_VARPER + img * K_SEG;
    for (int i = tid; i < K_SEG; i += blockDim.x)
        if (s_var[i] != 0.f) atomicAdd(&gvar[i], s_var[i]);
}

// Finalize: one wave32 per image. Gram = m @ m^T via V_WMMA_F32_16X16X4_F32.
__global__ void k_final(const float* __restrict__ ws, float* __restrict__ out) {
    __shared__ float s_m[NIMG][32][EDIM];   // means of instances 1..32
    __shared__ float s_diag[NIMG][32];      // ||m_i||^2 (Gram diagonal)
    __shared__ float s_cntv[NIMG][32];
    __shared__ int   s_pres[NIMG][32];
    __shared__ float s_acc[NIMG][4];        // var, dist, reg, n_inst
    __shared__ float s_res[NIMG][4];        // v, d, r, valid

    const int tid = threadIdx.x;
    for (int i = tid; i < NIMG * 32 * EDIM; i += blockDim.x) {
        int img = i >> 9, rem = i & 511, row = rem >> 4, e = rem & 15;
        s_m[img][row][e] = ws[OFF_MEANS + img * K_SEG * EDIM + (row + 1) * EDIM + e];
    }
    for (int i = tid; i < NIMG * 32; i += blockDim.x) {
        int img = i >> 5, row = i & 31;
        float c = ws[OFF_COUNTS + img * K_SEG + row + 1];
        s_pres[img][row] = (c > 0.f) ? 1 : 0;
        s_cntv[img][row] = fmaxf(c, 1.f);
    }
    if (tid < NIMG * 4) s_acc[tid >> 2][tid & 3] = 0.f;
    __syncthreads();

    const int wave = tid >> 5;
    const int lane = tid & 31;
    const int half = lane >> 4;    // lanes 16-31 hold K pair {2,3}
    const int lr   = lane & 15;
    const int img  = wave;

    // Gram tiles (0,0), (1,1), (0,1); K=16 accumulated as 4 chunks of 4.
    v8f acc00 = {}; v8f acc11 = {}; v8f acc01 = {};
    #pragma unroll
    for (int kk = 0; kk < 4; ++kk) {
        int ks = kk * 4 + half * 2;
        v2f a0, a1;
        a0.x = s_m[img][lr][ks];      a0.y = s_m[img][lr][ks + 1];
        a1.x = s_m[img][16 + lr][ks]; a1.y = s_m[img][16 + lr][ks + 1];
        acc00 = __builtin_amdgcn_wmma_f32_16x16x4_f32(false, a0, false, a0, (short)0, acc00, false, false);
        acc11 = __builtin_amdgcn_wmma_f32_16x16x4_f32(false, a1, false, a1, (short)0, acc11, false, false);
        acc01 = __builtin_amdgcn_wmma_f32_16x16x4_f32(false, a0, false, a1, (short)0, acc01, false, false);
    }

    // Diagonal extraction: D element r at lane L holds G[r + 8*(L>=16)][L&15].
    #pragma unroll
    for (int r = 0; r < 8; ++r) {
        int il = r + half * 8;
        if (il == lr) {
            s_diag[img][il]      = acc00[r];
            s_diag[img][16 + il] = acc11[r];
        }
    }
    __syncthreads();

    // Pairwise hinge distance accumulation (i < j only).
    float ldist = 0.f;
    #pragma unroll
    for (int r = 0; r < 8; ++r) {
        int i0 = r + half * 8;
        int j0 = lr;
        if (j0 > i0 && s_pres[img][i0] && s_pres[img][j0]) {
            float dsq = s_diag[img][i0] + s_diag[img][j0] - 2.f * acc00[r];
            float d = dsq > 0.f ? sqrtf(dsq) : 0.f;
            float h = fmaxf(2.f * DELTA_D - d, 0.f);
            ldist = fmaf(h, h, ldist);
        }
        if (j0 > i0 && s_pres[img][16 + i0] && s_pres[img][16 + j0]) {
            float dsq = s_diag[img][16 + i0] + s_diag[img][16 + j0] - 2.f * acc11[r];
            float d = dsq > 0.f ? sqrtf(dsq) : 0.f;
            float h = fmaxf(2.f * DELTA_D - d, 0.f);
            ldist = fmaf(h, h, ldist);
        }
        if (s_pres[img][i0] && s_pres[img][16 + j0]) {   // always i < j
            float dsq = s_diag[img][i0] + s_diag[img][16 + j0] - 2.f * acc01[r];
            float d = dsq > 0.f ? sqrtf(dsq) : 0.f;
            float h = fmaxf(2.f * DELTA_D - d, 0.f);
            ldist = fmaf(h, h, ldist);
        }
    }
    atomicAdd(&s_acc[img][1], ldist);

    // Per-instance stats: lane == instance index (0..31 -> segment 1..32).
    {
        float pres = s_pres[img][lane] ? 1.f : 0.f;
        float vvar = pres * (ws[OFF_VARPER + img * K_SEG + lane + 1] / s_cntv[img][lane]);
        float dg = s_diag[img][lane];
        float vreg = (pres != 0.f && dg > 0.f) ? sqrtf(dg) : 0.f;
        atomicAdd(&s_acc[img][0], vvar);
        atomicAdd(&s_acc[img][2], vreg);
        atomicAdd(&s_acc[img][3], pres);
    }
    __syncthreads();

    if (tid < NIMG) {
        float n   = s_acc[tid][3];
        float nin = fmaxf(n, 1.f);
        float v   = s_acc[tid][0] / nin;
        float npairs = fmaxf(n * (n - 1.f) * 0.5f, 1.f);
        float d   = (n > 1.f) ? s_acc[tid][1] / npairs : 0.f;
        float r   = s_acc[tid][2] / nin;
        float valid = (n > 0.f) ? 1.f : 0.f;
        s_res[tid][0] = v * valid;
        s_res[tid][1] = d * valid;
        s_res[tid][2] = r * valid;
        s_res[tid][3] = valid;
    }
    __syncthreads();

    if (tid == 0) {
        float sv = 0.f, sd = 0.f, sr = 0.f, sval = 0.f;
        for (int i = 0; i < NIMG; ++i) {
            sv += s_res[i][0]; sd += s_res[i][1]; sr += s_res[i][2]; sval += s_res[i][3];
        }
        float vs = fmaxf(sval, 1.f);
        float var_loss = sv / vs, dist_loss = sd / vs, reg_loss = sr / vs;
        out[0] = ALPHA * var_loss + BETA * dist_loss + GAMMA * reg_loss;
        out[1] = var_loss;
        out[2] = dist_loss;
        out[3] = reg_loss;
    }
}

extern "C" void kernel_launch(void* const* d_in, const int* in_sizes, int n_in,
                              void* d_out, int out_size, void* d_ws, size_t ws_size,
                              hipStream_t stream) {
    const float* emb  = (const float*)d_in[0];
    const int*   mask = (const int*)d_in[1];
    float* ws  = (float*)d_ws;
    float* out = (float*)d_out;

    k_zero<<<(WS_FLOATS + TPB - 1) / TPB, TPB, 0, stream>>>(ws);

    dim3 gstream(PIX_BLOCKS, NIMG);
    k_segsum<<<gstream, TPB, 0, stream>>>(emb, mask, ws);

    k_means<<<(NIMG * K_SEG * EDIM + TPB - 1) / TPB, TPB, 0, stream>>>(ws);

    k_var<<<gstream, TPB, 0, stream>>>(emb, mask, ws);

    k_final<<<1, TPB, 0, stream>>>(ws, out);
}